// CondPF_40553081209357
// MI455X (gfx1250) — compile-verified
//
#include <hip/hip_runtime.h>

// Conditional particle filter (CondPF), CDNA5/gfx1250.
// T=32 steps, N=32768 particles, dx=8, 4 Euler substeps/step.
// Propagation uses V_WMMA_F32_16X16X4_F32: per wave, a 16-particle tile is
// kept in the 16x16 f32 C/D accumulator layout; each substep chains 4 WMMAs
//   X' = X + dW*Sig^T + X*(h*A)^T       (K=8 split as 2 slices of K=4)
// with the X A-operand built via a per-wave LDS transpose and dW generated
// directly in A-operand layout from a counter RNG.
//
// Workspace layout (floats): P0, P1 (ping-pong (T+1,N,8) history, ~34.6MB
// each -> both L2-resident on the 192MB L2), gn[N], bins[N], scal[64],
// sel[N] (int). Total ~66.5 MiB.

#define TT   32
#define NP   32768
#define DXC  8
#define NSUB 4
#define HL       0.25f   // 2^-L, L=2
#define SQRT_HL  0.5f

typedef __attribute__((ext_vector_type(2))) float v2f;
typedef __attribute__((ext_vector_type(8))) float v8f;

// ---------------- counter-based RNG (splitmix64 + Box-Muller) ----------------
__device__ __forceinline__ unsigned long long mix64(unsigned long long x) {
  x += 0x9E3779B97F4A7C15ull;
  x = (x ^ (x >> 30)) * 0xBF58476D1CE4E5B9ull;
  x = (x ^ (x >> 27)) * 0x94D049BB133111EBull;
  return x ^ (x >> 31);
}

__device__ __forceinline__ v2f normal2(unsigned long long key) {
  unsigned long long r = mix64(key);
  unsigned int u1 = (unsigned int)r;
  unsigned int u2 = (unsigned int)(r >> 32);
  float f1 = ((float)u1 + 1.0f) * 2.3283064365386963e-10f;  // (0,1]
  float f2 = (float)u2 * 2.3283064365386963e-10f;           // [0,1)
  float rad = sqrtf(-2.0f * __logf(f1));
  float sn, cs;
  __sincosf(6.283185307179586f * f2, &sn, &cs);
  v2f o; o.x = rad * cs; o.y = rad * sn;
  return o;
}

__device__ __forceinline__ float uniform1(unsigned long long key) {
  return (float)(unsigned int)(mix64(key) >> 40) * (1.0f / 16777216.0f);
}

__device__ __forceinline__ int searchsorted_right(const float* __restrict__ bins, float v) {
  int lo = 0, hi = NP;
  while (lo < hi) {
    int mid = (lo + hi) >> 1;
    if (bins[mid] <= v) lo = mid + 1; else hi = mid;
  }
  return lo < NP - 1 ? lo : NP - 1;
}

// ---------------- init: row 0 of P0, gn=0, pin particle N-1 ----------------
__global__ __launch_bounds__(256)
void pf_init(float* __restrict__ P0, float* __restrict__ gn,
             const float* __restrict__ input_path) {
  int n = blockIdx.x * 256 + threadIdx.x;
  if (n >= NP) return;
  gn[n] = 0.0f;
#pragma unroll
  for (int d = 0; d < DXC; ++d)
    P0[(size_t)n * DXC + d] = (n == NP - 1) ? input_path[d] : 0.0f;
}

// ---------------- propagate: Euler-Maruyama via WMMA ----------------
// Each wave owns 16 particles. 8 waves/block, 256 blocks -> N particles.
__global__ __launch_bounds__(256)
void pf_propagate(float* __restrict__ P, const float* __restrict__ Amat,
                  const float* __restrict__ Sigm,
                  const float* __restrict__ input_path, int t) {
  __shared__ float xtile[8][16 * DXC];  // per-wave 16x8 transpose staging
  const int lane = threadIdx.x & 31;
  const int wave = threadIdx.x >> 5;
  const int sub  = lane & 15;
  const int half = lane >> 4;
  const int base = (blockIdx.x * 8 + wave) * 16;
  float* lds = xtile[wave];

  const float* __restrict__ Xrow = P + (size_t)t * NP * DXC;
  float* __restrict__ Yrow       = P + (size_t)(t + 1) * NP * DXC;

  // B operands: 4x16 slices of (h*A)^T and Sig^T, cols >= 8 zero-padded.
  // Layout: VGPR0 holds K-rows {0,2} (lanes 0-15 / 16-31), VGPR1 rows {1,3}.
  v2f bA_lo = {0.f, 0.f}, bA_hi = {0.f, 0.f};
  v2f bS_lo = {0.f, 0.f}, bS_hi = {0.f, 0.f};
  if (sub < DXC) {
    const int d = sub, k = 2 * half;
    bA_lo.x = HL * Amat[d * 8 + k];     bA_lo.y = HL * Amat[d * 8 + k + 1];
    bA_hi.x = HL * Amat[d * 8 + k + 4]; bA_hi.y = HL * Amat[d * 8 + k + 5];
    bS_lo.x = Sigm[d * 8 + k];          bS_lo.y = Sigm[d * 8 + k + 1];
    bS_hi.x = Sigm[d * 8 + k + 4];      bS_hi.y = Sigm[d * 8 + k + 5];
  }

  // X in WMMA C/D layout: lane (col d = sub, M-half = half), VGPR v = row m%8.
  v8f c = {0.f, 0.f, 0.f, 0.f, 0.f, 0.f, 0.f, 0.f};
  if (sub < DXC) {
#pragma unroll
    for (int v = 0; v < 8; ++v)
      c[v] = Xrow[(size_t)(base + v + 8 * half) * DXC + sub];
  }

  const int m = sub;  // A-operand row for this lane
#pragma unroll
  for (int s = 0; s < NSUB; ++s) {
    // C-layout -> LDS 16x8 tile (cols 8..15 are zero, not stored)
    if (sub < DXC) {
#pragma unroll
      for (int v = 0; v < 8; ++v)
        lds[(v + 8 * half) * DXC + sub] = c[v];
    }
    asm volatile("s_wait_dscnt 0x0" ::: "memory");

    // A operand for X: lane m, K = {2h,2h+1} (lo) / {4+2h,5+2h} (hi)
    v2f ax_lo, ax_hi;
    ax_lo.x = lds[m * DXC + 2 * half];
    ax_lo.y = lds[m * DXC + 2 * half + 1];
    ax_hi.x = lds[m * DXC + 2 * half + 4];
    ax_hi.y = lds[m * DXC + 2 * half + 5];

    // dW generated directly in A-operand layout (no shuffle needed)
    const unsigned long long kb =
        ((unsigned long long)(t * NSUB + s) << 20) |
        ((unsigned long long)(base + m) << 2);
    v2f aw_lo = normal2(kb | (unsigned long long)half);
    v2f aw_hi = normal2(kb | (unsigned long long)(2 + half));
    aw_lo.x *= SQRT_HL; aw_lo.y *= SQRT_HL;
    aw_hi.x *= SQRT_HL; aw_hi.y *= SQRT_HL;

    // X' = X + dW*Sig^T + X*(h*A)^T  (four chained 16x16x4 f32 WMMAs)
    c = __builtin_amdgcn_wmma_f32_16x16x4_f32(false, aw_lo, false, bS_lo,
                                              (short)0, c, false, false);
    c = __builtin_amdgcn_wmma_f32_16x16x4_f32(false, aw_hi, false, bS_hi,
                                              (short)0, c, false, false);
    c = __builtin_amdgcn_wmma_f32_16x16x4_f32(false, ax_lo, false, bA_lo,
                                              (short)0, c, false, false);
    c = __builtin_amdgcn_wmma_f32_16x16x4_f32(false, ax_hi, false, bA_hi,
                                              (short)0, c, false, false);
  }

  // Store row t+1; pin particle N-1 to input_path[t+1].
  if (sub < DXC) {
#pragma unroll
    for (int v = 0; v < 8; ++v) {
      const int p = base + v + 8 * half;
      float val = c[v];
      if (p == NP - 1) val = input_path[(size_t)(t + 1) * DXC + sub];
      Yrow[(size_t)p * DXC + sub] = val;
    }
  }
}

// ------- fused llg + softmax stats + ESS + cumsum (single workgroup) -------
__global__ __launch_bounds__(1024)
void pf_weights(const float* __restrict__ P, const float* __restrict__ observe_path,
                float* __restrict__ gn, float* __restrict__ bins,
                float* __restrict__ scal, int t) {
  __shared__ float red[1024];
  const int tid = threadIdx.x;
  const float* __restrict__ X  = P + (size_t)(t + 1) * NP * DXC;
  const float* __restrict__ yp = observe_path + (size_t)(t + 1) * DXC;
  const float y0 = yp[0], y1 = yp[1], y2 = yp[2], y3 = yp[3];
  const float y4 = yp[4], y5 = yp[5], y6 = yp[6], y7 = yp[7];

  // (1) gn += llg, track max
  float lmax = -3.4e38f;
  for (int i = tid; i < NP; i += 1024) {
    const float4* xp = (const float4*)(X + (size_t)i * DXC);
    float4 a = xp[0], b = xp[1];
    float s = (a.x - y0) * (a.x - y0) + (a.y - y1) * (a.y - y1) +
              (a.z - y2) * (a.z - y2) + (a.w - y3) * (a.w - y3) +
              (b.x - y4) * (b.x - y4) + (b.y - y5) * (b.y - y5) +
              (b.z - y6) * (b.z - y6) + (b.w - y7) * (b.w - y7);
    float g = gn[i] - 0.5f * s;
    gn[i] = g;
    lmax = fmaxf(lmax, g);
  }
  red[tid] = lmax; __syncthreads();
  for (int o = 512; o > 0; o >>= 1) {
    if (tid < o) red[tid] = fmaxf(red[tid], red[tid + o]);
    __syncthreads();
  }
  const float gmax = red[0]; __syncthreads();

  // (2) S = sum exp, S2 = sum exp^2
  float lS = 0.f, lS2 = 0.f;
  for (int i = tid; i < NP; i += 1024) {
    float e = __expf(gn[i] - gmax);
    lS += e; lS2 += e * e;
  }
  red[tid] = lS; __syncthreads();
  for (int o = 512; o > 0; o >>= 1) { if (tid < o) red[tid] += red[tid + o]; __syncthreads(); }
  const float S = red[0]; __syncthreads();
  red[tid] = lS2; __syncthreads();
  for (int o = 512; o > 0; o >>= 1) { if (tid < o) red[tid] += red[tid + o]; __syncthreads(); }
  const float S2 = red[0]; __syncthreads();

  const float sumw2 = S2 / (S * S);                 // = 1/ESS
  const int do_res = sumw2 >= (2.0f / (float)NP);   // ESS <= N/2
  if (tid == 0) { scal[0] = gmax; scal[1] = S; scal[2] = do_res ? 1.0f : 0.0f; }

  // (3) cumsum of w = exp(gn-gmax)/S : per-thread chunk of 32 + LDS scan
  const float invS = 1.0f / S;
  const int cbase = tid * 32;
  float csum = 0.f;
  for (int j = 0; j < 32; ++j) {
    csum += __expf(gn[cbase + j] - gmax) * invS;
    bins[cbase + j] = csum;
  }
  red[tid] = csum; __syncthreads();
  for (int o = 1; o < 1024; o <<= 1) {
    float v = (tid >= o) ? red[tid - o] : 0.f;
    __syncthreads();
    red[tid] += v;
    __syncthreads();
  }
  const float excl = red[tid] - csum;
  for (int j = 0; j < 32; ++j) bins[cbase + j] += excl;

  // (4) reset gn on resample
  if (do_res)
    for (int j = 0; j < 32; ++j) gn[cbase + j] = 0.f;
}

// ---------------- selection: multinomial ancestor index ----------------
__global__ __launch_bounds__(256)
void pf_sel(const float* __restrict__ bins, const float* __restrict__ scal,
            int* __restrict__ sel, int t) {
  int n = blockIdx.x * 256 + threadIdx.x;
  if (n >= NP) return;
  int s = n;  // identity when no resampling
  if (scal[2] > 0.5f) {
    float dice = uniform1(0x5EEDB00Cull + ((unsigned long long)t << 32) +
                          (unsigned long long)n);
    s = searchsorted_right(bins, dice);
  }
  sel[n] = s;
}

// ------- ancestry gather: dst[r][n] = src[r][sel[n]], rows 0..t+1 -------
__global__ __launch_bounds__(256)
void pf_gather(const float* __restrict__ src, float* __restrict__ dst,
               const int* __restrict__ sel, const float* __restrict__ input_path,
               int rows) {
  int gid = blockIdx.x * 256 + threadIdx.x;  // one float4 per thread
  int total = rows * NP * 2;
  if (gid >= total) return;
  const int h  = gid & 1;
  const int rn = gid >> 1;
  const int n  = rn & (NP - 1);
  const int r  = rn >> 15;  // log2(NP)
  float4 val;
  if (n == NP - 1) {
    val = ((const float4*)(input_path + (size_t)r * DXC))[h];  // pinned path
  } else {
    val = ((const float4*)(src + ((size_t)r * NP + sel[n]) * DXC))[h];
  }
  ((float4*)(dst + ((size_t)r * NP + n) * DXC))[h] = val;
}

// ---------------- final: sample one index from wn, emit its path ----------------
__global__ __launch_bounds__(256)
void pf_final(const float* __restrict__ src, const float* __restrict__ bins,
              float* __restrict__ out) {
  __shared__ int sh_idx;
  if (threadIdx.x == 0) {
    float dice = uniform1(0xF01DAB1Eull ^ 0x77ull);
    sh_idx = searchsorted_right(bins, dice);
  }
  __syncthreads();
  const int idx = sh_idx;
  for (int i = threadIdx.x; i < (TT + 1) * DXC; i += 256) {
    const int r = i >> 3, d = i & 7;
    out[i] = src[((size_t)r * NP + idx) * DXC + d];
  }
}

// ---------------- host-side launch pipeline ----------------
extern "C" void kernel_launch(void* const* d_in, const int* in_sizes, int n_in,
                              void* d_out, int out_size, void* d_ws, size_t ws_size,
                              hipStream_t stream) {
  (void)in_sizes; (void)n_in; (void)out_size; (void)ws_size;
  const float* input_path   = (const float*)d_in[0];
  const float* observe_path = (const float*)d_in[1];
  const float* Amat         = (const float*)d_in[2];
  const float* Sigm         = (const float*)d_in[3];
  float* out = (float*)d_out;

  float* ws = (float*)d_ws;
  const size_t PATHF = (size_t)(TT + 1) * NP * DXC;
  float* P0   = ws;
  float* P1   = P0 + PATHF;
  float* gn   = P1 + PATHF;
  float* bins = gn + NP;
  float* scal = bins + NP;
  int*   sel  = (int*)(scal + 64);

  pf_init<<<NP / 256, 256, 0, stream>>>(P0, gn, input_path);

  for (int t = 0; t < TT; ++t) {
    float* src = (t & 1) ? P1 : P0;
    float* dst = (t & 1) ? P0 : P1;

    pf_propagate<<<NP / 128, 256, 0, stream>>>(src, Amat, Sigm, input_path, t);
    pf_weights<<<1, 1024, 0, stream>>>(src, observe_path, gn, bins, scal, t);

    if (t < TT - 1) {
      pf_sel<<<NP / 256, 256, 0, stream>>>(bins, scal, sel, t);
      const int rows  = t + 2;
      const int total = rows * NP * 2;
      pf_gather<<<(total + 255) / 256, 256, 0, stream>>>(src, dst, sel,
                                                         input_path, rows);
    }
  }

  // t=31 propagated into P1 (pre-resample state = reference's final `un`).
  pf_final<<<1, 256, 0, stream>>>(P1, bins, out);
}